// WeightedRuleLayer_60464549593339
// MI455X (gfx1250) — compile-verified
//
#include <hip/hip_runtime.h>
#include <stdint.h>

// Problem constants (fixed by the reference).
#define D_DIM 16
#define FANIN 4
#define RULES_PER_BLOCK 64
#define THREADS (RULES_PER_BLOCK * 4)            // 256 threads = 8 wave32; 4 lanes per rule
#define IDX_PER_BLOCK (RULES_PER_BLOCK * FANIN)  // 256 dwords = 1 KB of indices per block

typedef float    f32x4 __attribute__((ext_vector_type(4)));
typedef uint32_t u32x4 __attribute__((ext_vector_type(4)));
typedef uint32_t u32x8 __attribute__((ext_vector_type(8)));

__device__ __forceinline__ float fast_tanh(float x) {
  // tanh(x) = 1 - 2/(exp(2x)+1); exp(2x) = exp2(x * 2*log2(e))
  // Saturates to +/-1 for large |x| with no inf/inf NaN. v_exp_f32 + v_rcp_f32.
  float e = __builtin_amdgcn_exp2f(x * 2.885390081777927f);
  return 1.0f - 2.0f * __builtin_amdgcn_rcpf(e + 1.0f);
}

__global__ __launch_bounds__(THREADS) void WeightedRuleLayer_kernel(
    const float* __restrict__ layer_values,  // [N_SRC, 16]
    const int*   __restrict__ indices,       // [B*4]
    const float* __restrict__ weights,       // [4, 16]
    float*       __restrict__ out,           // [B, 16]
    int B)
{
  __shared__ int s_idx[IDX_PER_BLOCK];

  const uint32_t idx_base  = (uint32_t)blockIdx.x * IDX_PER_BLOCK;
  const uint32_t total_idx = (uint32_t)B * FANIN;

#if defined(__AMDGCN__)
  // Wave 0 issues one TDM DMA of this block's index slab (global -> LDS).
  // TDM ignores EXEC; the branch is wave-uniform so exactly one issue/block.
  if (threadIdx.x < 32u) {
    const uint64_t gaddr = (uint64_t)(uintptr_t)(indices + idx_base);
    const uint32_t dim0  = total_idx - idx_base;  // valid elements from tile start (OOB reads -> 0)

    // D# group 0 (ISA 8.3): count=1 | lds_addr | global_addr[56:0] | type=2
    u32x4 g0;
    g0.x = 1u;                                              // count=1, user descriptor
    g0.y = 0u;                                              // lds_addr: s_idx at block LDS base
    g0.z = (uint32_t)gaddr;                                 // global_addr[31:0]
    g0.w = (uint32_t)((gaddr >> 32) & 0x01FFFFFFu)          // global_addr[56:32]
         | (2u << 30);                                      // type=2 ("image")

    // D# group 1 (ISA 8.4): mask=0, data_size=4B, dims/strides for a 1-row 256-dword tile
    u32x8 g1;
    g1.s0 = 2u << 16;                                       // data_size=2 (4 bytes); no flags
    g1.s1 = (dim0 & 0xFFFFu) << 16;                         // tensor_dim0[15:0]  -> bits[63:48]
    g1.s2 = (dim0 >> 16) | (1u << 16);                      // tensor_dim0[31:16]; tensor_dim1=1
    g1.s3 = (uint32_t)IDX_PER_BLOCK << 16;                  // tile_dim0=256      -> bits[127:112]
    g1.s4 = 0u;                                             // tile_dim1=0, tile_dim2=0 (unused)
    g1.s5 = (uint32_t)IDX_PER_BLOCK;                        // tensor_dim0_stride (single row)
    g1.s6 = 0u;
    g1.s7 = 0u;

    // 2-operand form: groups 2/3 NULL (tensor <= 2D). Portable inline-asm path.
    asm volatile("tensor_load_to_lds %0, %1" :: "s"(g0), "s"(g1) : "memory");
    __builtin_amdgcn_s_wait_tensorcnt(0);
  }
#endif
  __syncthreads();

  const int  r   = threadIdx.x >> 2;   // rule within block
  const int  sub = threadIdx.x & 3;    // which float4 of the 16-float row
  const long b   = (long)blockIdx.x * RULES_PER_BLOCK + r;
  if (b >= B) return;

  const f32x4* __restrict__ lv = (const f32x4*)layer_values;  // row n = lv[n*4 .. n*4+3]
  const f32x4* __restrict__ wq = (const f32x4*)weights;       // wq[i*4 + sub]

  f32x4 acc = (f32x4)0.0f;
#pragma unroll
  for (int i = 0; i < FANIN; ++i) {
    const int   n = s_idx[r * FANIN + i];
    const f32x4 x = lv[(long)n * 4 + sub];   // quad of lanes covers the full 64B row
    const f32x4 w = wq[i * 4 + sub];         // broadcast across block (256B total, L1-resident)
    acc += w * x;                            // diag(w) @ x == elementwise scale
  }

  f32x4 res;
  res.x = fast_tanh(acc.x);
  res.y = fast_tanh(acc.y);
  res.z = fast_tanh(acc.z);
  res.w = fast_tanh(acc.w);

  // Output is write-once (128 MB): non-temporal store keeps L2 reserved for
  // layer_values (128 MB, fully L2-resident -> cheap gathers).
  __builtin_nontemporal_store(res, (f32x4*)out + b * 4 + sub);
}

extern "C" void kernel_launch(void* const* d_in, const int* in_sizes, int n_in,
                              void* d_out, int out_size, void* d_ws, size_t ws_size,
                              hipStream_t stream) {
  const float* layer_values = (const float*)d_in[0];
  const int*   indices      = (const int*)d_in[1];
  const float* weights      = (const float*)d_in[2];
  float*       out          = (float*)d_out;

  const int B      = in_sizes[1] / FANIN;
  const int blocks = (B + RULES_PER_BLOCK - 1) / RULES_PER_BLOCK;

  WeightedRuleLayer_kernel<<<blocks, THREADS, 0, stream>>>(
      layer_values, indices, weights, out, B);
}